// SGNN_30855045054720
// MI455X (gfx1250) — compile-verified
//
#include <hip/hip_runtime.h>
#include <hip/hip_bf16.h>

#define NN   8192     // num nodes == feature dim
#define H1C  128
#define H2C  64
#define NE   262144
#define NB   1024

typedef __attribute__((ext_vector_type(2))) float v2f;
typedef __attribute__((ext_vector_type(8))) float v8f;

// ---------------------------------------------------------------------------
// GEMM1: h1 = relu(X @ W1^T), duplicated into agg1 (self-loop seed).
// Block = 256 threads = 8 waves covering a 32-row x 128-col output macro-tile:
//   wave w: M-tile = w>>2 (2 x 16 rows), col pair = w&3 (2 x 16 cols).
// Each wave computes TWO 16x16 tiles so each A (X) fragment feeds 2 WMMAs:
// 3 b64 loads per 2 WMMAs instead of 4. X (256 MB) is the HBM-bound stream;
// W1 (4 MB) stays L2-resident.
// A (16x4 f32): lane<16 -> row l, K={k0,k0+1}; lane>=16 -> K={k0+2,k0+3}.
// B (4x16 f32): B[k, c] = W1[c, k]  -> per-lane contiguous pair from W1 row.
// ---------------------------------------------------------------------------
__global__ __launch_bounds__(256) void gemm1_kernel(const float* __restrict__ X,
                                                    const float* __restrict__ W1,
                                                    float* __restrict__ h1,
                                                    float* __restrict__ agg1) {
    const int lane  = threadIdx.x & 31;
    const int wave  = threadIdx.x >> 5;
    const int row0  = blockIdx.x * 32 + (wave >> 2) * 16;
    const int col0  = (wave & 3) * 32;
    const int lhalf = lane >> 4;            // 0 or 1
    const int lmod  = lane & 15;

    const float* Arow  = X  + (size_t)(row0 + lmod) * NN + 2 * lhalf;
    const float* Brow0 = W1 + (size_t)(col0 + lmod) * NN + 2 * lhalf;
    const float* Brow1 = W1 + (size_t)(col0 + 16 + lmod) * NN + 2 * lhalf;

    v8f c0 = {};
    v8f c1 = {};
#pragma unroll 4
    for (int k0 = 0; k0 < NN; k0 += 4) {
        v2f a  = *(const v2f*)(Arow + k0);
        v2f b0 = *(const v2f*)(Brow0 + k0);
        v2f b1 = *(const v2f*)(Brow1 + k0);
        c0 = __builtin_amdgcn_wmma_f32_16x16x4_f32(false, a, false, b0,
                                                   (short)0, c0, false, false);
        c1 = __builtin_amdgcn_wmma_f32_16x16x4_f32(false, a, false, b1,
                                                   (short)0, c1, false, false);
    }

#pragma unroll
    for (int v = 0; v < 8; ++v) {
        int    r  = row0 + v + 8 * lhalf;   // C/D: VGPR v -> M = v (+8 for hi half)
        size_t b  = (size_t)r * H1C + lmod; //      N = lane % 16
        float  v0 = fmaxf(c0[v], 0.0f);
        float  v1 = fmaxf(c1[v], 0.0f);
        h1  [b + col0]      = v0;
        agg1[b + col0]      = v0;
        h1  [b + col0 + 16] = v1;
        agg1[b + col0 + 16] = v1;
    }
}

// ---------------------------------------------------------------------------
// GEMM2: h2 = relu(agg1 @ W2^T), duplicated into rep. M=8192, N=64, K=128.
// One block = one 16-row tile, 4 waves = 4 x 16-col tiles (64 cols).
// ---------------------------------------------------------------------------
__global__ __launch_bounds__(128) void gemm2_kernel(const float* __restrict__ agg1,
                                                    const float* __restrict__ W2,
                                                    float* __restrict__ h2,
                                                    float* __restrict__ rep) {
    const int lane  = threadIdx.x & 31;
    const int wave  = threadIdx.x >> 5;     // col tile 0..3
    const int row0  = blockIdx.x * 16;
    const int col0  = wave * 16;
    const int lhalf = lane >> 4;
    const int lmod  = lane & 15;

    const float* Arow = agg1 + (size_t)(row0 + lmod) * H1C + 2 * lhalf;
    const float* Brow = W2   + (size_t)(col0 + lmod) * H1C + 2 * lhalf;

    v8f c = {};
#pragma unroll
    for (int k0 = 0; k0 < H1C; k0 += 4) {
        v2f a = *(const v2f*)(Arow + k0);
        v2f b = *(const v2f*)(Brow + k0);
        c = __builtin_amdgcn_wmma_f32_16x16x4_f32(false, a, false, b,
                                                  (short)0, c, false, false);
    }

#pragma unroll
    for (int v = 0; v < 8; ++v) {
        int   r   = row0 + v + 8 * lhalf;
        int   cc  = col0 + lmod;
        float val = fmaxf(c[v], 0.0f);
        h2 [(size_t)r * H2C + cc] = val;
        rep[(size_t)r * H2C + cc] = val;
    }
}

// ---------------------------------------------------------------------------
// Edge scatter-add: accum[dst] += feat[src]  (accum pre-seeded with feat = self loop)
// C/4 threads per edge, float4 gather + 4 f32 atomics.
// ---------------------------------------------------------------------------
template <int C>
__global__ __launch_bounds__(256) void scatter_add_kernel(const float* __restrict__ feat,
                                                          float* __restrict__ accum,
                                                          const int* __restrict__ esrc,
                                                          const int* __restrict__ edst,
                                                          int nE) {
    constexpr int TPE = C / 4;
    long long t = (long long)blockIdx.x * blockDim.x + threadIdx.x;
    long long e = t / TPE;
    if (e >= nE) return;
    int ch = (int)(t % TPE) * 4;

    int s = esrc[e];
    int d = edst[e];
    const float4 v = *(const float4*)(feat + (size_t)s * C + ch);
    float* p = accum + (size_t)d * C + ch;
    atomicAdd(p + 0, v.x);
    atomicAdd(p + 1, v.y);
    atomicAdd(p + 2, v.z);
    atomicAdd(p + 3, v.w);
}

// ---------------------------------------------------------------------------
// Loss: sum |rep[nodes] @ rep[nodes]^T - D[nodes][:,nodes]|.
// One wave per 16x32 tile of S (A-fragment reused across 2 WMMAs),
// node-indirect row pointers; wave32 shuffle reduction + 1 atomic per wave.
// ---------------------------------------------------------------------------
__global__ __launch_bounds__(32) void loss_kernel(const float* __restrict__ rep,
                                                  const float* __restrict__ D,
                                                  const int* __restrict__ nodes,
                                                  float* __restrict__ out) {
    const int lane  = threadIdx.x & 31;
    const int lhalf = lane >> 4;
    const int lmod  = lane & 15;
    const int row0  = blockIdx.y * 16;
    const int col0  = blockIdx.x * 32;

    const float* Arow  = rep + (size_t)nodes[row0 + lmod] * H2C + 2 * lhalf;
    const float* Brow0 = rep + (size_t)nodes[col0 + lmod] * H2C + 2 * lhalf;
    const float* Brow1 = rep + (size_t)nodes[col0 + 16 + lmod] * H2C + 2 * lhalf;

    v8f c0 = {};
    v8f c1 = {};
#pragma unroll
    for (int k0 = 0; k0 < H2C; k0 += 4) {
        v2f a  = *(const v2f*)(Arow + k0);
        v2f b0 = *(const v2f*)(Brow0 + k0);
        v2f b1 = *(const v2f*)(Brow1 + k0);
        c0 = __builtin_amdgcn_wmma_f32_16x16x4_f32(false, a, false, b0,
                                                   (short)0, c0, false, false);
        c1 = __builtin_amdgcn_wmma_f32_16x16x4_f32(false, a, false, b1,
                                                   (short)0, c1, false, false);
    }

    float sum = 0.0f;
#pragma unroll
    for (int v = 0; v < 8; ++v) {
        int    r   = row0 + v + 8 * lhalf;
        size_t dr  = (size_t)nodes[r] * NN;
        float  d0  = D[dr + nodes[col0 + lmod]];
        float  d1  = D[dr + nodes[col0 + 16 + lmod]];
        sum += fabsf(c0[v] - d0);
        sum += fabsf(c1[v] - d1);
    }
    // wave32 tree reduction
#pragma unroll
    for (int m = 16; m >= 1; m >>= 1) sum += __shfl_xor(sum, m, 32);
    if (lane == 0) atomicAdd(out, sum);
}

__global__ void zero_kernel(float* __restrict__ out) {
    if (threadIdx.x == 0 && blockIdx.x == 0) out[0] = 0.0f;
}

// ---------------------------------------------------------------------------
extern "C" void kernel_launch(void* const* d_in, const int* in_sizes, int n_in,
                              void* d_out, int out_size, void* d_ws, size_t ws_size,
                              hipStream_t stream) {
    const float* X     = (const float*)d_in[0];   // [NN, NN]
    const float* W1    = (const float*)d_in[1];   // [H1C, NN]
    const float* W2    = (const float*)d_in[2];   // [H2C, H1C]
    const float* D     = (const float*)d_in[3];   // [NN, NN]
    const int*   esrc  = (const int*)d_in[4];     // [NE]
    const int*   edst  = (const int*)d_in[5];     // [NE]
    const int*   nodes = (const int*)d_in[6];     // [NB]
    float*       out   = (float*)d_out;

    float* ws   = (float*)d_ws;
    float* h1   = ws;                             // NN*H1C
    float* agg1 = h1   + (size_t)NN * H1C;        // NN*H1C
    float* h2   = agg1 + (size_t)NN * H1C;        // NN*H2C
    float* rep  = h2   + (size_t)NN * H2C;        // NN*H2C  (12 MB total)

    zero_kernel<<<1, 32, 0, stream>>>(out);

    gemm1_kernel<<<NN / 32, 256, 0, stream>>>(X, W1, h1, agg1);

    {
        long long threads = (long long)NE * (H1C / 4);
        int blocks = (int)((threads + 255) / 256);
        scatter_add_kernel<H1C><<<blocks, 256, 0, stream>>>(h1, agg1, esrc, edst, NE);
    }

    gemm2_kernel<<<NN / 16, 128, 0, stream>>>(agg1, W2, h2, rep);

    {
        long long threads = (long long)NE * (H2C / 4);
        int blocks = (int)((threads + 255) / 256);
        scatter_add_kernel<H2C><<<blocks, 256, 0, stream>>>(h2, rep, esrc, edst, NE);
    }

    dim3 g(NB / 32, NB / 16);
    loss_kernel<<<g, 32, 0, stream>>>(rep, D, nodes, out);
}